// SSMLayerFFTComplex_10230612099619
// MI455X (gfx1250) — compile-verified
//
#include <hip/hip_runtime.h>
#include <hip/hip_bf16.h>

typedef __attribute__((ext_vector_type(2))) float v2f;
typedef __attribute__((ext_vector_type(8))) float v8f;

// Problem constants (from reference setup_inputs)
#define B_     8
#define INCH   32
#define OUTCH  32
#define T_     4096
#define P_     128        // num_pairs (complex modes; conjugates folded via 2*Re)
#define CHUNK  64
#define NCHUNK (T_ / CHUNK)   // 64

// Workspace layout (float offsets)
#define PARAM_OFF 0                       // 6*P_ floats: A_re,A_im,f_re,f_im,AL_re,AL_im
#define V_OFF     1024
#define V_SIZE    ((size_t)B_ * T_ * P_)  // 4,194,304 floats (16 MB)
#define E_OFF     (V_OFF + V_SIZE)
#define E_SIZE    ((size_t)B_ * NCHUNK * P_ * 2)  // 131072 floats
#define C_OFF     (E_OFF + E_SIZE)
#define X_OFF     (C_OFF + E_SIZE)        // another V_SIZE floats (2*Re(x), laid out (b,t,p))

// ---------------------------------------------------------------------------
// Phase 0: per-pair discretization.  lam = -softplus(raw_lambda) + i*raw_omega
// A_d = exp(lam);  factor = (A_d-1)/lam (or DT=1 if |lam|<=EPS);  AL = A_d^CHUNK
// ---------------------------------------------------------------------------
__global__ void ssm_setup(const float* __restrict__ rl,
                          const float* __restrict__ ro,
                          float* __restrict__ ws) {
    int p = threadIdx.x;
    if (p >= P_) return;
    float x  = rl[p];
    float sp = fmaxf(x, 0.0f) + log1pf(expf(-fabsf(x)));   // stable softplus
    float lre = -sp;
    float lim = ro[p];
    float er  = expf(lre);
    float Are = er * cosf(lim);
    float Aim = er * sinf(lim);
    float mag2 = lre * lre + lim * lim;
    float fre, fim;
    if (sqrtf(mag2) > 1e-6f) {
        // (A-1)/lam = ((Are-1) + i*Aim) * (lre - i*lim) / |lam|^2
        fre = ((Are - 1.0f) * lre + Aim * lim) / mag2;
        fim = (Aim * lre - (Are - 1.0f) * lim) / mag2;
    } else {
        fre = 1.0f;  // DT
        fim = 0.0f;
    }
    // A_d^64 by 6 complex squarings
    float pr = Are, pi = Aim;
    #pragma unroll
    for (int i = 0; i < 6; ++i) {
        float nr = pr * pr - pi * pi;
        float ni = 2.0f * pr * pi;
        pr = nr; pi = ni;
    }
    ws[PARAM_OFF + 0 * P_ + p] = Are;
    ws[PARAM_OFF + 1 * P_ + p] = Aim;
    ws[PARAM_OFF + 2 * P_ + p] = fre;
    ws[PARAM_OFF + 3 * P_ + p] = fim;
    ws[PARAM_OFF + 4 * P_ + p] = pr;
    ws[PARAM_OFF + 5 * P_ + p] = pi;
}

// ---------------------------------------------------------------------------
// Phase 1 (WMMA): V[b,t,p] = sum_i u[b,i,t] * B_c[i,p]
// One wave per (b, 16-t tile, 16-pair tile); K=32 via 8 steps of 16x16x4 f32.
// A fragment: A[m=t, k=i]; B fragment: B[k=i, n=pair].
// ---------------------------------------------------------------------------
__global__ void ssm_gemm1(const float* __restrict__ u,
                          const float* __restrict__ Bc,
                          float* __restrict__ ws) {
    int wave = blockIdx.x * (blockDim.x >> 5) + (threadIdx.x >> 5);
    int lane = threadIdx.x & 31;
    int pt = wave & 7;           // pair tile  (8)
    int tt = (wave >> 3) & 255;  // t tile     (256)
    int b  = wave >> 11;         // batch      (8)
    int half = lane >> 4;        // ISA: lanes 16-31 hold K=2,3
    int m    = lane & 15;
    int t0 = tt * 16, p0 = pt * 16;
    const float* ub = u + (size_t)b * INCH * T_;

    v8f acc = {};
    #pragma unroll
    for (int kk = 0; kk < INCH; kk += 4) {
        int k0 = kk + 2 * half;
        v2f a, bb;
        a.x  = ub[(size_t)(k0    ) * T_ + t0 + m];
        a.y  = ub[(size_t)(k0 + 1) * T_ + t0 + m];
        bb.x = Bc[(k0    ) * P_ + p0 + m];
        bb.y = Bc[(k0 + 1) * P_ + p0 + m];
        acc = __builtin_amdgcn_wmma_f32_16x16x4_f32(
            false, a, false, bb, (short)0, acc, false, false);
    }
    float* V = ws + V_OFF + (size_t)b * T_ * P_;
    #pragma unroll
    for (int r = 0; r < 8; ++r) {
        int t = t0 + r + 8 * half;           // C layout: lanes>=16 hold M=r+8
        V[(size_t)t * P_ + p0 + m] = acc[r];
    }
}

// ---------------------------------------------------------------------------
// Phase 2a: per-chunk local scan ends (zero init): e_c = local x at chunk end.
// Lanes map 32 consecutive pairs -> coalesced 128B reads of V each t-step.
// ---------------------------------------------------------------------------
__global__ void ssm_scan_ends(float* __restrict__ ws) {
    int tid = blockIdx.x * blockDim.x + threadIdx.x;   // 65536 total
    int p = tid & (P_ - 1);
    int c = (tid >> 7) & (NCHUNK - 1);
    int b = tid >> 13;
    float Are = ws[PARAM_OFF + 0 * P_ + p], Aim = ws[PARAM_OFF + 1 * P_ + p];
    float fre = ws[PARAM_OFF + 2 * P_ + p], fim = ws[PARAM_OFF + 3 * P_ + p];
    const float* V = ws + V_OFF + ((size_t)b * T_ + (size_t)c * CHUNK) * P_ + p;
    float xr = 0.0f, xi = 0.0f;
    #pragma unroll 4
    for (int t = 0; t < CHUNK; ++t) {
        float v  = V[(size_t)t * P_];
        float nr = fmaf(Are, xr, fmaf(-Aim, xi, fre * v));
        float ni = fmaf(Are, xi, fmaf( Aim, xr, fim * v));
        xr = nr; xi = ni;
    }
    size_t idx = ((size_t)(b * NCHUNK + c) * P_ + p) * 2;
    ws[E_OFF + idx]     = xr;
    ws[E_OFF + idx + 1] = xi;
}

// ---------------------------------------------------------------------------
// Phase 2b: stitch chunk carries.  carry_{c+1} = A^CHUNK * carry_c + e_c.
// Stores carry-IN for each chunk.  Only 1024 threads x 64 steps.
// ---------------------------------------------------------------------------
__global__ void ssm_stitch(float* __restrict__ ws) {
    int tid = blockIdx.x * blockDim.x + threadIdx.x;   // 1024 total
    int p = tid & (P_ - 1);
    int b = tid >> 7;
    float ALr = ws[PARAM_OFF + 4 * P_ + p], ALi = ws[PARAM_OFF + 5 * P_ + p];
    float cr = 0.0f, ci = 0.0f;
    for (int c = 0; c < NCHUNK; ++c) {
        size_t idx = ((size_t)(b * NCHUNK + c) * P_ + p) * 2;
        ws[C_OFF + idx]     = cr;
        ws[C_OFF + idx + 1] = ci;
        float er = ws[E_OFF + idx], ei = ws[E_OFF + idx + 1];
        float nr = fmaf(ALr, cr, fmaf(-ALi, ci, er));
        float ni = fmaf(ALr, ci, fmaf( ALi, cr, ei));
        cr = nr; ci = ni;
    }
}

// ---------------------------------------------------------------------------
// Phase 2c: re-scan each chunk with known carry-in; emit 2*Re(x) (conjugate
// pair folding) into Xr laid out (b, t, p).
// ---------------------------------------------------------------------------
__global__ void ssm_rescan(float* __restrict__ ws) {
    int tid = blockIdx.x * blockDim.x + threadIdx.x;
    int p = tid & (P_ - 1);
    int c = (tid >> 7) & (NCHUNK - 1);
    int b = tid >> 13;
    float Are = ws[PARAM_OFF + 0 * P_ + p], Aim = ws[PARAM_OFF + 1 * P_ + p];
    float fre = ws[PARAM_OFF + 2 * P_ + p], fim = ws[PARAM_OFF + 3 * P_ + p];
    size_t cidx = ((size_t)(b * NCHUNK + c) * P_ + p) * 2;
    float xr = ws[C_OFF + cidx];
    float xi = ws[C_OFF + cidx + 1];
    const float* V = ws + V_OFF + ((size_t)b * T_ + (size_t)c * CHUNK) * P_ + p;
    float*      X = ws + X_OFF + ((size_t)b * T_ + (size_t)c * CHUNK) * P_ + p;
    #pragma unroll 4
    for (int t = 0; t < CHUNK; ++t) {
        float v  = V[(size_t)t * P_];
        float nr = fmaf(Are, xr, fmaf(-Aim, xi, fre * v));
        float ni = fmaf(Are, xi, fmaf( Aim, xr, fim * v));
        xr = nr; xi = ni;
        X[(size_t)t * P_] = 2.0f * xr;   // 2*Re(x): conjugate-pair fold
    }
}

// ---------------------------------------------------------------------------
// Phase 3 (WMMA): y[b,o,t] = sum_p Xr2[b,t,p] * C_pair[p,o]
// One wave per (b, 16-t tile, 16-o tile); K=128 via 32 steps of 16x16x4 f32.
// ---------------------------------------------------------------------------
__global__ void ssm_gemm2(const float* __restrict__ Cp,
                          const float* __restrict__ ws,
                          float* __restrict__ y) {
    int wave = blockIdx.x * (blockDim.x >> 5) + (threadIdx.x >> 5);
    int lane = threadIdx.x & 31;
    int ot = wave & 1;           // o tile (2)
    int tt = (wave >> 1) & 255;  // t tile (256)
    int b  = wave >> 9;          // batch  (8)
    int half = lane >> 4;
    int m    = lane & 15;
    int t0 = tt * 16, o0 = ot * 16;
    const float* X = ws + X_OFF + (size_t)b * T_ * P_;

    v8f acc = {};
    #pragma unroll
    for (int kk = 0; kk < P_; kk += 4) {
        int k0 = kk + 2 * half;
        v2f a, bb;
        const float* row = X + (size_t)(t0 + m) * P_;
        a.x  = row[k0];
        a.y  = row[k0 + 1];
        bb.x = Cp[(k0    ) * OUTCH + o0 + m];
        bb.y = Cp[(k0 + 1) * OUTCH + o0 + m];
        acc = __builtin_amdgcn_wmma_f32_16x16x4_f32(
            false, a, false, bb, (short)0, acc, false, false);
    }
    #pragma unroll
    for (int r = 0; r < 8; ++r) {
        int t = t0 + r + 8 * half;
        y[((size_t)b * OUTCH + o0 + m) * T_ + t] = acc[r];
    }
}

// ---------------------------------------------------------------------------
extern "C" void kernel_launch(void* const* d_in, const int* in_sizes, int n_in,
                              void* d_out, int out_size, void* d_ws, size_t ws_size,
                              hipStream_t stream) {
    const float* u  = (const float*)d_in[0];  // (8, 32, 4096)
    const float* rl = (const float*)d_in[1];  // (128,)
    const float* ro = (const float*)d_in[2];  // (128,)
    const float* Bc = (const float*)d_in[3];  // (32, 128)
    const float* Cp = (const float*)d_in[4];  // (128, 32)
    float* y  = (float*)d_out;                // (8, 32, 4096)
    float* ws = (float*)d_ws;                 // needs ~35 MB

    ssm_setup    <<<1,    128, 0, stream>>>(rl, ro, ws);
    ssm_gemm1    <<<4096, 128, 0, stream>>>(u, Bc, ws);   // 16384 waves
    ssm_scan_ends<<<256,  256, 0, stream>>>(ws);          // 65536 threads
    ssm_stitch   <<<4,    256, 0, stream>>>(ws);          // 1024 threads
    ssm_rescan   <<<256,  256, 0, stream>>>(ws);
    ssm_gemm2    <<<1024, 128, 0, stream>>>(Cp, ws, y);   // 4096 waves
}